// ModelNew_25056839205050
// MI455X (gfx1250) — compile-verified
//
#include <hip/hip_runtime.h>
#include <stdint.h>

#define C_CH     2048
#define GRP_SZ   32
#define CH_PER_T 8
#define BLK      256
#define EPS_GN   1e-5f

typedef float    f4    __attribute__((ext_vector_type(4)));
typedef uint32_t u32x4 __attribute__((ext_vector_type(4)));
typedef int      i32x8 __attribute__((ext_vector_type(8)));
typedef int      i32x4 __attribute__((ext_vector_type(4)));

// Butterfly add across lanes via LDS-swizzle (group-of-32 mode, and_mask=0x1f):
// 0x041F = xor 1, 0x081F = xor 2  -> full sum over each aligned 4-lane subgroup.
#define SWZ_ADD(v, imm) \
  ((v) + __int_as_float(__builtin_amdgcn_ds_swizzle(__float_as_int(v), (imm))))

// bias-add -> hardtanh[-1,1] -> fast mish (matches reference approximations).
// hardtanh via single v_med3_f32; softplus output is in [0, ~1.32] so the
// reference's clamp(sp, -5, 5) is a provable no-op and is elided.
__device__ __forceinline__ float ht_mish(float xv) {
  float v  = __builtin_amdgcn_fmed3f(xv, -1.0f, 1.0f);
  float a  = __builtin_fabsf(v);
  float e  = __expf(-a);                               // v_exp_f32 path
  float sp = fmaxf(v, 0.0f) + __logf(1.0f + e);        // v_log_f32 path
  float s2 = sp * sp;
  float t  = __fdividef(sp * (27.0f + s2),             // rcp + mul fast divide
                        fmaf(9.0f, s2, 27.0f));
  return v * t;
}

#if __has_builtin(__builtin_amdgcn_tensor_load_to_lds)
#define USE_TDM 1
// 1-D TDM copy: nelem f32 elements from global -> LDS, per ISA D# layout.
// This toolchain uses the 6-arg builtin (g0 v4u, g1 v8i, g2 v4i, g3 v4i,
// extra v8i, cpol i32).
__device__ __forceinline__ void tdm_load_1d(uint64_t gaddr, uint32_t lds_byte,
                                            int nelem) {
  u32x4 g0;
  g0[0] = 1u;                                            // count=1, user D#
  g0[1] = lds_byte;                                      // LDS dest (bytes)
  g0[2] = (uint32_t)gaddr;                               // global addr [31:0]
  g0[3] = ((uint32_t)(gaddr >> 32) & 0x01FFFFFFu)        // global addr [56:32]
          | (2u << 30);                                  // type = 2 ("image")
  i32x8 g1;
  g1[0] = (int)(2u << 16);                               // data_size = 4 bytes
  g1[1] = (int)(((uint32_t)nelem & 0xFFFFu) << 16);      // tensor_dim0 [15:0]
  g1[2] = (int)((((uint32_t)nelem >> 16) & 0xFFFFu)      // tensor_dim0 [31:16]
          | (1u << 16));                                 // tensor_dim1 = 1
  g1[3] = (int)(((uint32_t)nelem & 0xFFFFu) << 16);      // tile_dim0
  g1[4] = 0;                                             // tile_dim1/2 unused
  g1[5] = nelem;                                         // tensor_dim0_stride
  g1[6] = 0;
  g1[7] = 0;
  i32x4 z4 = {0, 0, 0, 0};
  i32x8 z8 = {0, 0, 0, 0, 0, 0, 0, 0};
  __builtin_amdgcn_tensor_load_to_lds(g0, g1, z4, z4, z8, 0);
}
#else
#define USE_TDM 0
#endif

__global__ __launch_bounds__(BLK) void gn_mish_kernel(
    const float* __restrict__ x, const float* __restrict__ bias,
    const float* __restrict__ gamma, const float* __restrict__ beta,
    float* __restrict__ out, int nrows) {
  const int tid   = threadIdx.x;
  const int cbase = tid * CH_PER_T;

#if USE_TDM
  __shared__ __align__(16) float smem[3 * C_CH];
  if (tid == 0) {
    // TDM EXEC-agnostic; issue once from wave 0, other waves branch around.
    uint32_t lb = (uint32_t)(uintptr_t)&smem[0];         // flat lo32 = LDS off
    tdm_load_1d((uint64_t)(uintptr_t)bias,  lb,                C_CH);
    tdm_load_1d((uint64_t)(uintptr_t)gamma, lb + 4u * C_CH,    C_CH);
    tdm_load_1d((uint64_t)(uintptr_t)beta,  lb + 8u * C_CH,    C_CH);
    __builtin_amdgcn_s_wait_tensorcnt(0);
  }
  __syncthreads();
  const float* bp = smem;
  const float* gp = smem + C_CH;
  const float* ep = smem + 2 * C_CH;
#else
  const float* bp = bias;
  const float* gp = gamma;
  const float* ep = beta;
#endif

  // Register-cache this thread's 8 channels of bias/gamma/beta for all rows.
  float pb[CH_PER_T], pg[CH_PER_T], pe[CH_PER_T];
  {
    f4 b0 = *(const f4*)(bp + cbase); f4 b1 = *(const f4*)(bp + cbase + 4);
    f4 g0 = *(const f4*)(gp + cbase); f4 g1 = *(const f4*)(gp + cbase + 4);
    f4 e0 = *(const f4*)(ep + cbase); f4 e1 = *(const f4*)(ep + cbase + 4);
#pragma unroll
    for (int i = 0; i < 4; ++i) {
      pb[i] = b0[i]; pb[4 + i] = b1[i];
      pg[i] = g0[i]; pg[4 + i] = g1[i];
      pe[i] = e0[i]; pe[4 + i] = e1[i];
    }
  }

  for (int row = blockIdx.x; row < nrows; row += gridDim.x) {
    const float* xr = x + (size_t)row * C_CH + cbase;
    f4 x0 = *(const f4*)(xr);
    f4 x1 = *(const f4*)(xr + 4);

    int nr = row + gridDim.x;
    if (nr < nrows)
      __builtin_prefetch(x + (size_t)nr * C_CH + cbase, 0, 1);

    float v[CH_PER_T];
#pragma unroll
    for (int i = 0; i < 4; ++i) {
      v[i]     = ht_mish(x0[i] + pb[i]);
      v[4 + i] = ht_mish(x1[i] + pb[4 + i]);
    }

    // Per-thread partials over 8 channels, then 4-lane butterfly -> group of 32.
    float s = 0.0f, q = 0.0f;
#pragma unroll
    for (int i = 0; i < CH_PER_T; ++i) {
      s += v[i];
      q  = fmaf(v[i], v[i], q);
    }
    s = SWZ_ADD(s, 0x041F); s = SWZ_ADD(s, 0x081F);
    q = SWZ_ADD(q, 0x041F); q = SWZ_ADD(q, 0x081F);

    const float mean = s * (1.0f / GRP_SZ);
    const float var  = fmaf(q, 1.0f / GRP_SZ, -mean * mean);
    const float inv  = __builtin_amdgcn_rsqf(var + EPS_GN);

    f4 o0, o1;
#pragma unroll
    for (int i = 0; i < 4; ++i) {
      o0[i] = fmaf((v[i]     - mean) * inv, pg[i],     pe[i]);
      o1[i] = fmaf((v[4 + i] - mean) * inv, pg[4 + i], pe[4 + i]);
    }
    float* orow = out + (size_t)row * C_CH + cbase;
    *(f4*)(orow)     = o0;
    *(f4*)(orow + 4) = o1;
  }
}

extern "C" void kernel_launch(void* const* d_in, const int* in_sizes, int n_in,
                              void* d_out, int out_size, void* d_ws,
                              size_t ws_size, hipStream_t stream) {
  (void)n_in; (void)out_size; (void)d_ws; (void)ws_size;
  const float* x     = (const float*)d_in[0];
  const float* bias  = (const float*)d_in[1];
  const float* gamma = (const float*)d_in[2];
  const float* beta  = (const float*)d_in[3];
  float* out = (float*)d_out;

  const int nrows = in_sizes[0] / C_CH;  // 32768 for the reference shapes
  int grid = 4096;                       // persistent blocks, ~8 rows each
  if (grid > nrows) grid = nrows;

  hipLaunchKernelGGL(gn_mish_kernel, dim3(grid), dim3(BLK), 0, stream,
                     x, bias, gamma, beta, out, nrows);
}